// MaxUnpooling2D_38568806318557
// MI455X (gfx1250) — compile-verified
//
#include <hip/hip_runtime.h>
#include <stdint.h>

// MaxUnpooling2D scatter-add for fixed shapes:
//   updates: [16, 64, 64, 256] f32   (2^24 elements, per-batch 2^20)
//   mask:    [16, 64, 64, 256] i32   values in [0, 2^22) = flat idx into [128,128,256]
//   out:     [16, 128, 128, 256] f32 (2^26 elements, per-batch 2^22)
// out_idx = (b << 22) | (mask & ~0xFF) | c      (all dims are powers of two)

#define THREADS        256
#define VEC            4                    // elements per lane per tile (one b128)
#define TILE           (THREADS * VEC)      // 1024 elements per tile
#define TILES_PER_BLK  16                   // 16384 elements per block

__global__ void unpool_zero_kernel(float4* __restrict__ out4, int n4) {
    int i = blockIdx.x * blockDim.x + threadIdx.x;
    if (i < n4) {
        out4[i] = make_float4(0.f, 0.f, 0.f, 0.f);   // RT stores: keep lines warm for atomics
    }
}

__global__ __launch_bounds__(THREADS)
void unpool_scatter_kernel(const float* __restrict__ upd,
                           const int*   __restrict__ msk,
                           float*       __restrict__ out,
                           int n) {
    // Double-buffered LDS staging: 2 x (4KB updates + 4KB mask) = 16 KB.
    __shared__ __align__(16) float s_upd[2][TILE];
    __shared__ __align__(16) int   s_msk[2][TILE];

    const int tid       = threadIdx.x;
    const int laneBase  = tid * VEC;                    // element offset of this lane in tile
    const int blockBase = blockIdx.x * (TILES_PER_BLK * TILE);

    // Issue one tile's worth of async global->LDS copies (2 instructions / wave).
    // NT hint: inputs are streamed exactly once; don't evict output lines from L2.
    auto issue = [&](int buf, int t) {
        int g = blockBase + t * TILE + laneBase;        // element index (16B-aligned bytes)
        if (g + VEC > n) g = 0;                         // clamp: staged but never consumed
        uint64_t gu = (uint64_t)(uintptr_t)(upd + g);
        uint64_t gm = (uint64_t)(uintptr_t)(msk + g);
        uint32_t lu = (uint32_t)(uintptr_t)&s_upd[buf][laneBase];
        uint32_t lm = (uint32_t)(uintptr_t)&s_msk[buf][laneBase];
        asm volatile("global_load_async_to_lds_b128 %0, %1, off th:TH_LOAD_NT"
                     :: "v"(lu), "v"(gu) : "memory");
        asm volatile("global_load_async_to_lds_b128 %0, %1, off th:TH_LOAD_NT"
                     :: "v"(lm), "v"(gm) : "memory");
    };

    issue(0, 0);                                        // prime the pipeline

    for (int t = 0; t < TILES_PER_BLK; ++t) {
        const int buf = t & 1;
        if (t + 1 < TILES_PER_BLK) {
            issue((t + 1) & 1, t + 1);                  // prefetch next tile
            asm volatile("s_wait_asynccnt 2" ::: "memory");  // tile t landed in LDS
        } else {
            asm volatile("s_wait_asynccnt 0" ::: "memory");
        }

        const int g = blockBase + t * TILE + laneBase;  // this lane's 4 elements
        if (g + VEC <= n) {
            const uint32_t bBits = ((uint32_t)(g >> 20)) << 22;  // batch -> output-slice base
            const uint32_t cBase = (uint32_t)g & 255u;           // channel of element 0
#pragma unroll
            for (int k = 0; k < VEC; ++k) {
                float    v = s_upd[buf][laneBase + k];
                uint32_t m = (uint32_t)s_msk[buf][laneBase + k];
                // (y*Wo + x)*C == m & ~0xFF ; channel comes from position, not mask.
                uint32_t idx = bBits | (m & 0xFFFFFF00u) | (cBase + (uint32_t)k);
                uint32_t off = idx << 2;                // byte offset into out (< 2^28)
                // No-return f32 atomic add at device scope: RMW at the L2 atomic units,
                // coherent across all workgroups (duplicate indices collide cross-block).
                asm volatile("global_atomic_add_f32 %0, %1, %2 scope:SCOPE_DEV"
                             :: "v"(off), "v"(v), "s"(out) : "memory");
            }
        }
    }
}

extern "C" void kernel_launch(void* const* d_in, const int* in_sizes, int n_in,
                              void* d_out, int out_size, void* d_ws, size_t ws_size,
                              hipStream_t stream) {
    const float* upd = (const float*)d_in[0];
    const int*   msk = (const int*)d_in[1];
    float*       out = (float*)d_out;

    const int n  = in_sizes[0];       // 2^24 input elements
    const int n4 = out_size >> 2;     // 2^24 float4 stores to zero 2^26 floats

    // 1) Zero the output (harness poisons it; atomics require a zero base).
    unpool_zero_kernel<<<(n4 + THREADS - 1) / THREADS, THREADS, 0, stream>>>(
        (float4*)out, n4);

    // 2) Async-staged, double-buffered scatter-add.
    const int tiles  = (n + TILE - 1) / TILE;
    const int blocks = (tiles + TILES_PER_BLK - 1) / TILES_PER_BLK;   // 1024 blocks
    unpool_scatter_kernel<<<blocks, THREADS, 0, stream>>>(upd, msk, out, n);
}